// MyModel_82497731822305
// MI455X (gfx1250) — compile-verified
//
#include <hip/hip_runtime.h>
#include <hip/hip_bf16.h>

#define T_LEN 16384
#define NBAND 64
#define KW    129   // taps
#define KPAD  132   // padded to multiple of 4
#define HALF_K 64   // (KW-1)/2, 'SAME' padding offset

#define BLK_T   64          // time samples per block in kernel 1
#define WIN_LEN 208         // 64 + 128 halo, rounded up (max used index = 194)

typedef __attribute__((ext_vector_type(2))) float v2f;
typedef __attribute__((ext_vector_type(8))) float v8f;
typedef __attribute__((ext_vector_type(4))) unsigned int v4u;
typedef __attribute__((ext_vector_type(8))) int v8i;
typedef __attribute__((ext_vector_type(4))) int v4i;

#if defined(__has_builtin)
#if __has_builtin(__builtin_amdgcn_tensor_load_to_lds)
#define HAVE_TDM 1
#endif
#endif

#ifdef HAVE_TDM
// ---------------------------------------------------------------------------
// Issue one TDM load: a (tile1 x tile0) fp32 tile from a row-major tensor
// (dims dim1 x dim0, row stride stride0 elements) into contiguous LDS.
// All arguments are wave-uniform -> descriptor built in SGPRs.
// Reads past tensor_dim return zero.
// ---------------------------------------------------------------------------
__device__ inline void tdm_load_2d_f32(const void* gaddr, unsigned lds_addr,
                                       unsigned dim0, unsigned dim1,
                                       unsigned tile0, unsigned tile1,
                                       unsigned stride0)
{
    unsigned long long ga = (unsigned long long)gaddr;
    v4u g0;
    g0[0] = 1u;                                           // count=1, user D#
    g0[1] = lds_addr;                                     // LDS byte address
    g0[2] = (unsigned)(ga & 0xffffffffu);                 // global_addr[31:0]
    g0[3] = (unsigned)((ga >> 32) & 0x01ffffffu)          // global_addr[56:32]
            | 0x80000000u;                                // type=2 (image)
    v8i g1;
    g1[0] = 0x00020000;                                   // data_size=4B, no mcast
    g1[1] = (int)((dim0 & 0xffffu) << 16);                // tensor_dim0[15:0]
    g1[2] = (int)((dim0 >> 16) | ((dim1 & 0xffffu) << 16));
    g1[3] = (int)((dim1 >> 16) | (tile0 << 16));          // tile_dim0
    g1[4] = (int)(tile1 & 0xffffu);                       // tile_dim1 (tile_dim2=0)
    g1[5] = (int)stride0;                                 // tensor_dim0_stride[31:0]
    g1[6] = 0;
    g1[7] = 0;
    v4i z = {};
#if __clang_major__ >= 23
    v8i z8 = {};
    __builtin_amdgcn_tensor_load_to_lds(g0, g1, z, z, z8, 0);
#else
    __builtin_amdgcn_tensor_load_to_lds(g0, g1, z, z, 0);
#endif
}

__device__ inline void wait_tensorcnt0() {
#if __has_builtin(__builtin_amdgcn_s_wait_tensorcnt)
    __builtin_amdgcn_s_wait_tensorcnt(0);
#else
    asm volatile("s_wait_tensorcnt 0x0" ::: "memory");
#endif
}
#endif // HAVE_TDM

// ---------------------------------------------------------------------------
// Kernel 1: complex conv as GEMM via V_WMMA_F32_16X16X4_F32.
//   x1[b,t] = sum_k w[b,k] * x[t + k - 64]   (complex)
// Block = 4 waves; each wave owns one 16(M=bands) x 16(N=time) tile.
// TDM stages the block's x window (with halo) and the 16x129 weight tile
// into LDS; the 32 steady-state K-steps then run guard-free from ds_load.
// ---------------------------------------------------------------------------
__global__ __launch_bounds__(128) void conv_wmma_kernel(
    const float* __restrict__ xr, const float* __restrict__ xi,
    const float* __restrict__ wr, const float* __restrict__ wi,
    float* __restrict__ x1r, float* __restrict__ x1i)
{
    __shared__ float s_xr[WIN_LEN], s_xi[WIN_LEN];
    __shared__ float s_wr[16 * KW], s_wi[16 * KW];

    const int lane = threadIdx.x & 31;
    const int wave = threadIdx.x >> 5;

    const int mb     = blockIdx.x;            // band tile 0..3
    const int tb     = blockIdx.y;            // 64-wide time block 0..255
    const int t0_blk = tb * BLK_T;

    // ---- Stage LDS tiles -------------------------------------------------
#ifdef HAVE_TDM
    {
        const int shift  = (tb == 0) ? HALF_K : 0;        // left 'SAME' pad
        const int tstart = t0_blk - HALF_K + shift;
        const int navail = T_LEN - tstart;
        const int want   = WIN_LEN - shift;
        const int nload  = (want < navail) ? want : navail;

        // zero left pad (block 0) and right tail (last block); the TDM
        // span is clamped to T (nload <= navail), so no read passes T.
        for (int i = threadIdx.x; i < WIN_LEN; i += 128) {
            if (i < shift || i >= shift + nload) { s_xr[i] = 0.0f; s_xi[i] = 0.0f; }
        }

        if (wave == 0) {   // wave-uniform: only wave 0 issues the 4 TDM ops
            const unsigned axr = (unsigned)(unsigned long long)(void*)s_xr + shift * 4u;
            const unsigned axi = (unsigned)(unsigned long long)(void*)s_xi + shift * 4u;
            tdm_load_2d_f32(xr + tstart, axr, (unsigned)navail, 1u,
                            (unsigned)nload, 1u, (unsigned)navail);
            tdm_load_2d_f32(xi + tstart, axi, (unsigned)navail, 1u,
                            (unsigned)nload, 1u, (unsigned)navail);
            const unsigned awr = (unsigned)(unsigned long long)(void*)s_wr;
            const unsigned awi = (unsigned)(unsigned long long)(void*)s_wi;
            tdm_load_2d_f32(wr + mb * 16 * KW, awr, KW, 16u, KW, 16u, KW);
            tdm_load_2d_f32(wi + mb * 16 * KW, awi, KW, 16u, KW, 16u, KW);
            wait_tensorcnt0();
        }
        __syncthreads();
    }
#else
    {
        for (int i = threadIdx.x; i < WIN_LEN; i += 128) {
            const int t = t0_blk - HALF_K + i;
            const bool v = (t >= 0) && (t < T_LEN);
            s_xr[i] = v ? xr[t] : 0.0f;
            s_xi[i] = v ? xi[t] : 0.0f;
        }
        for (int i = threadIdx.x; i < 16 * KW; i += 128) {
            s_wr[i] = wr[mb * 16 * KW + i];
            s_wi[i] = wi[mb * 16 * KW + i];
        }
        __syncthreads();
    }
#endif

    // ---- WMMA accumulation ----------------------------------------------
    const int lrow   = lane & 15;             // A: local band row, B: N col
    const int khalf  = (lane >> 4) * 2;       // lane half selects K pair
    const int tloc0  = wave * 16 + lrow;      // B: window-local base index

    v8f acc_rr = {}; // Wr * Xr
    v8f acc_ii = {}; // Wi * Xi
    v8f acc_ri = {}; // Wr * Xi
    v8f acc_ir = {}; // Wi * Xr

    // 32 guard-free steps (k = 0..127)
    for (int ks = 0; ks < 32; ++ks) {
        const int k0 = ks * 4 + khalf;
        v2f ar, ai, br, bi;
#pragma unroll
        for (int j = 0; j < 2; ++j) {
            ar[j] = s_wr[lrow * KW + k0 + j];
            ai[j] = s_wi[lrow * KW + k0 + j];
            br[j] = s_xr[tloc0 + k0 + j];
            bi[j] = s_xi[tloc0 + k0 + j];
        }
        acc_rr = __builtin_amdgcn_wmma_f32_16x16x4_f32(
            false, ar, false, br, (short)0, acc_rr, false, false);
        acc_ii = __builtin_amdgcn_wmma_f32_16x16x4_f32(
            false, ai, false, bi, (short)0, acc_ii, false, false);
        acc_ri = __builtin_amdgcn_wmma_f32_16x16x4_f32(
            false, ar, false, bi, (short)0, acc_ri, false, false);
        acc_ir = __builtin_amdgcn_wmma_f32_16x16x4_f32(
            false, ai, false, br, (short)0, acc_ir, false, false);
    }
    // tail step (k = 128..131): zero A past KW; B may be anything (x * 0)
    {
        const int k0 = 128 + khalf;
        v2f ar, ai, br, bi;
#pragma unroll
        for (int j = 0; j < 2; ++j) {
            const int k = k0 + j;
            const bool kv = (k < KW);
            ar[j] = kv ? s_wr[lrow * KW + k] : 0.0f;
            ai[j] = kv ? s_wi[lrow * KW + k] : 0.0f;
            br[j] = s_xr[tloc0 + k];
            bi[j] = s_xi[tloc0 + k];
        }
        acc_rr = __builtin_amdgcn_wmma_f32_16x16x4_f32(
            false, ar, false, br, (short)0, acc_rr, false, false);
        acc_ii = __builtin_amdgcn_wmma_f32_16x16x4_f32(
            false, ai, false, bi, (short)0, acc_ii, false, false);
        acc_ri = __builtin_amdgcn_wmma_f32_16x16x4_f32(
            false, ar, false, bi, (short)0, acc_ri, false, false);
        acc_ir = __builtin_amdgcn_wmma_f32_16x16x4_f32(
            false, ai, false, br, (short)0, acc_ir, false, false);
    }

    // C/D layout: VGPR r, lanes 0-15 -> M = r, lanes 16-31 -> M = r + 8
    const int mrow = mb * 16 + (lane >> 4) * 8;
    const int col  = t0_blk + wave * 16 + (lane & 15);
#pragma unroll
    for (int r = 0; r < 8; ++r) {
        const int idx = (mrow + r) * T_LEN + col;
        x1r[idx] = acc_rr[r] - acc_ii[r];   // Re(w)*Re(x) - Im(w)*Im(x)
        x1i[idx] = acc_ri[r] + acc_ir[r];   // Re(w)*Im(x) + Im(w)*Re(x)
    }
}

// ---------------------------------------------------------------------------
// Kernel 2: RBF mixture + complex modulation. One block = one band x 256 t.
// Band coefficients staged in LDS (sigma pre-negated); 64 centers, 2 exps
// each -> trans-pipe bound (v_exp_f32), the true bottleneck of this net.
// ---------------------------------------------------------------------------
__global__ __launch_bounds__(256) void rbf_mod_kernel(
    const float* __restrict__ x1r, const float* __restrict__ x1i,
    const float* __restrict__ c_r, const float* __restrict__ c_i,
    const float* __restrict__ s_r, const float* __restrict__ s_i,
    const float* __restrict__ f_r, const float* __restrict__ f_i,
    float* __restrict__ x2r, float* __restrict__ x2i)
{
    __shared__ float lcr[NBAND], lci[NBAND], lsr[NBAND],
                     lsi[NBAND], lfr[NBAND], lfi[NBAND];

    const int b = blockIdx.x;
    const int t = blockIdx.y * 256 + threadIdx.x;

    if (threadIdx.x < NBAND) {
        const int c = threadIdx.x;
        lcr[c] = c_r[b * NBAND + c];
        lci[c] = c_i[b * NBAND + c];
        lsr[c] = -s_r[b * NBAND + c];   // pre-negate: exp(d*d*(-sigma))
        lsi[c] = -s_i[b * NBAND + c];
        lfr[c] = f_r[b * NBAND + c];
        lfi[c] = f_i[b * NBAND + c];
    }
    __syncthreads();

    const float vr = x1r[b * T_LEN + t];
    const float vi = x1i[b * T_LEN + t];
    const float a  = sqrtf(vr * vr + vi * vi);

    float outr = 0.0f, outi = 0.0f;
#pragma unroll 4
    for (int c = 0; c < NBAND; ++c) {
        const float dr = a - lcr[c];
        const float di = a - lci[c];
        outr += __expf(dr * dr * lsr[c]) * lfr[c];
        outi += __expf(di * di * lsi[c]) * lfi[c];
    }

    x2r[b * T_LEN + t] = outr * vr - outi * vi + vr;
    x2i[b * T_LEN + t] = outr * vi + outi * vr + vi;
}

// ---------------------------------------------------------------------------
// Kernel 3: reduce over bands with complex w3; interleaved complex64 output.
// ---------------------------------------------------------------------------
__global__ __launch_bounds__(256) void band_reduce_kernel(
    const float* __restrict__ x2r, const float* __restrict__ x2i,
    const float* __restrict__ w3r, const float* __restrict__ w3i,
    float* __restrict__ out)
{
    const int t = blockIdx.x * 256 + threadIdx.x;
    float yr = 0.0f, yi = 0.0f;
#pragma unroll 8
    for (int b = 0; b < NBAND; ++b) {
        const float r  = x2r[b * T_LEN + t];
        const float i  = x2i[b * T_LEN + t];
        const float wrv = w3r[b];   // uniform -> scalar loads
        const float wiv = w3i[b];
        yr += r * wrv - i * wiv;
        yi += r * wiv + i * wrv;
    }
    out[2 * t]     = yr;
    out[2 * t + 1] = yi;
}

extern "C" void kernel_launch(void* const* d_in, const int* in_sizes, int n_in,
                              void* d_out, int out_size, void* d_ws, size_t ws_size,
                              hipStream_t stream) {
    const float* xr  = (const float*)d_in[0];
    const float* xi  = (const float*)d_in[1];
    const float* w1r = (const float*)d_in[2];
    const float* w1i = (const float*)d_in[3];
    const float* cr  = (const float*)d_in[4];
    const float* ci  = (const float*)d_in[5];
    const float* sr  = (const float*)d_in[6];
    const float* si  = (const float*)d_in[7];
    const float* fr  = (const float*)d_in[8];
    const float* fi  = (const float*)d_in[9];
    const float* w3r = (const float*)d_in[10];
    const float* w3i = (const float*)d_in[11];
    float* out = (float*)d_out;

    const size_t plane = (size_t)NBAND * T_LEN;   // 1M floats = 4 MB
    float* ws  = (float*)d_ws;
    float* x1r = ws + 0 * plane;
    float* x1i = ws + 1 * plane;
    float* x2r = ws + 2 * plane;
    float* x2i = ws + 3 * plane;

    // Kernel 1: 4 band tiles x 256 time blocks x 4 waves
    conv_wmma_kernel<<<dim3(NBAND / 16, T_LEN / BLK_T), 128, 0, stream>>>(
        xr, xi, w1r, w1i, x1r, x1i);

    // Kernel 2: 64 bands x 64 time blocks
    rbf_mod_kernel<<<dim3(NBAND, T_LEN / 256), 256, 0, stream>>>(
        x1r, x1i, cr, ci, sr, si, fr, fi, x2r, x2i);

    // Kernel 3: final band reduction
    band_reduce_kernel<<<dim3(T_LEN / 256), 256, 0, stream>>>(
        x2r, x2i, w3r, w3i, out);
}